// SPDBatchNormLie_38508676776613
// MI455X (gfx1250) — compile-verified
//
#include <hip/hip_runtime.h>

// SPD BatchNorm (Lie / AIM metric) for B=4096, N=64, fp32.
// All matrix functions (sqrt, log, exp) implemented as matmul-only iterations
// so the whole pipeline runs on V_WMMA_F32_16X16X4_F32.

#define NMAT   64
#define NN     4096      // 64*64
#define NT     256       // threads per workgroup (8 wave32)
#define NWAVES 8
#define BATCH  4096
#define EPS_F  1e-5f
#define ALPHA_F 1.0f
#define BETA_F  0.0f
#define NS_ITERS 15      // Newton-Schulz iterations per sqrt
#define LOG_SQ   4       // inverse-scaling sqrt steps in matlog
#define LOG_ORD  8       // Mercator series order
#define EXP_SC   5       // scaling-squaring steps in matexp
#define EXP_ORD  8       // Taylor order

typedef float v2f __attribute__((ext_vector_type(2)));
typedef float v8f __attribute__((ext_vector_type(8)));

#define SH_MAT(name) __shared__ __align__(16) float name[NN]

// ---------------- workgroup 64x64 matmul: C = A * B (row-major, LDS) -------
// 16 output tiles of 16x16; 8 waves; wave w owns tiles (ti,tj) and (ti+32,tj)
// which share tj -> B fragments are loaded ONCE and feed two independent
// WMMA accumulator chains (halves LDS traffic, pipelines the XDL).
// A-fragment (16x4 f32): lane l: m=l%16, k=2*(l/16)+{0,1} -> contiguous b64.
// B-fragment (4x16 f32): lane l: n=l%16, k=2*(l/16)+{0,1} -> two b32, 256B
//   apart off one base (fusable to ds_load_2addr_stride64_b32).
// C/D (16x16 f32, 8 vgprs): vgpr g: row = g + 8*(l/16), col = l%16.
__device__ __forceinline__ void wg_mm(float* __restrict__ C,
                                      const float* __restrict__ A,
                                      const float* __restrict__ B) {
  __syncthreads();
  const int lane = threadIdx.x & 31;
  const int wave = threadIdx.x >> 5;
  const int m16  = lane & 15;
  const int half = lane >> 4;
  const int ti   = (wave >> 2) << 4;            // 0 or 16 (+32 for 2nd tile)
  const int tj   = (wave & 3) << 4;
  const float* a0p = A + (ti + m16) * NMAT + (half << 1);
  const float* a1p = a0p + 32 * NMAT;
  const float* bp  = B + (half << 1) * NMAT + tj + m16;
  v8f acc0 = {}, acc1 = {};
#pragma unroll
  for (int k4 = 0; k4 < 16; ++k4) {
    v2f av0 = *(const v2f*)(a0p + (k4 << 2));
    v2f av1 = *(const v2f*)(a1p + (k4 << 2));
    v2f bv;
    bv.x = bp[(k4 << 8)];
    bv.y = bp[(k4 << 8) + NMAT];
    acc0 = __builtin_amdgcn_wmma_f32_16x16x4_f32(false, av0, false, bv,
                                                 (short)0, acc0, false, false);
    acc1 = __builtin_amdgcn_wmma_f32_16x16x4_f32(false, av1, false, bv,
                                                 (short)0, acc1, false, false);
  }
  float* c0 = C + (ti + (half << 3)) * NMAT + tj + m16;
  float* c1 = c0 + 32 * NMAT;
#pragma unroll
  for (int g = 0; g < 8; ++g) {
    c0[g * NMAT] = acc0[g];
    c1[g * NMAT] = acc1[g];
  }
  __syncthreads();
}

// C = A * B^T  (B^T fragment is contiguous in B's rows -> b64 loads)
__device__ __forceinline__ void wg_mm_nt(float* __restrict__ C,
                                         const float* __restrict__ A,
                                         const float* __restrict__ B) {
  __syncthreads();
  const int lane = threadIdx.x & 31;
  const int wave = threadIdx.x >> 5;
  const int m16  = lane & 15;
  const int half = lane >> 4;
  const int ti   = (wave >> 2) << 4;
  const int tj   = (wave & 3) << 4;
  const float* a0p = A + (ti + m16) * NMAT + (half << 1);
  const float* a1p = a0p + 32 * NMAT;
  const float* bp  = B + (tj + m16) * NMAT + (half << 1);   // B^T[k][n]=B[n][k]
  v8f acc0 = {}, acc1 = {};
#pragma unroll
  for (int k4 = 0; k4 < 16; ++k4) {
    v2f av0 = *(const v2f*)(a0p + (k4 << 2));
    v2f av1 = *(const v2f*)(a1p + (k4 << 2));
    v2f bv  = *(const v2f*)(bp  + (k4 << 2));
    acc0 = __builtin_amdgcn_wmma_f32_16x16x4_f32(false, av0, false, bv,
                                                 (short)0, acc0, false, false);
    acc1 = __builtin_amdgcn_wmma_f32_16x16x4_f32(false, av1, false, bv,
                                                 (short)0, acc1, false, false);
  }
  float* c0 = C + (ti + (half << 3)) * NMAT + tj + m16;
  float* c1 = c0 + 32 * NMAT;
#pragma unroll
  for (int g = 0; g < 8; ++g) {
    c0[g * NMAT] = acc0[g];
    c1[g * NMAT] = acc1[g];
  }
  __syncthreads();
}

// ---------------- elementwise helpers (no internal barriers) ---------------
__device__ __forceinline__ void wg_copy(float* D, const float* S) {
  for (int e = threadIdx.x; e < NN; e += NT) D[e] = S[e];
}
__device__ __forceinline__ void wg_scale(float* D, const float* S, float a) {
  for (int e = threadIdx.x; e < NN; e += NT) D[e] = a * S[e];
}
__device__ __forceinline__ void wg_add_scaled(float* D, const float* S, float a) {
  for (int e = threadIdx.x; e < NN; e += NT) D[e] += a * S[e];
}
// D = a*S + bdiag*I   (D==S allowed)
__device__ __forceinline__ void wg_axpbI(float* D, const float* S, float a, float bdiag) {
  for (int e = threadIdx.x; e < NN; e += NT) {
    float v = a * S[e];
    if ((e >> 6) == (e & 63)) v += bdiag;
    D[e] = v;
  }
}
__device__ __forceinline__ void wg_identity(float* D) {
  for (int e = threadIdx.x; e < NN; e += NT)
    D[e] = ((e >> 6) == (e & 63)) ? 1.0f : 0.0f;
}
// in-place symmetrize (one writer per (i,j) pair)
__device__ __forceinline__ void wg_sym(float* D) {
  __syncthreads();
  for (int e = threadIdx.x; e < NN; e += NT) {
    int i = e >> 6, j = e & 63;
    if (i < j) {
      float v = 0.5f * (D[e] + D[j * NMAT + i]);
      D[e] = v;
      D[j * NMAT + i] = v;
    }
  }
  __syncthreads();
}
__device__ __forceinline__ float wg_trace(const float* S, float* red) {
  __syncthreads();
  if (threadIdx.x == 0) {
    float s = 0.0f;
    for (int i = 0; i < NMAT; ++i) s += S[i * (NMAT + 1)];
    red[0] = s;
  }
  __syncthreads();
  return red[0];
}

// ---------------- Newton-Schulz coupled sqrt: Y=sqrt(A), Z=invsqrt(A) ------
__device__ void wg_sqrt_ns(float* Y, float* Z, const float* A,
                           float* T, float* U, float* V, float* red) {
  float c  = wg_trace(A, red);           // trace bounds spectral radius (SPD)
  float ic = 1.0f / c;
  wg_scale(Y, A, ic);                    // Y0 = A/c  (eigs in (0,1])
  wg_identity(Z);                        // Z0 = I
#pragma clang loop unroll(disable)
  for (int it = 0; it < NS_ITERS; ++it) {
    wg_mm(T, Z, Y);                      // T = Z*Y
    wg_axpbI(T, T, -0.5f, 1.5f);         // T = (3I - Z*Y)/2
    wg_mm(U, Y, T);                      // Y' = Y*T
    wg_mm(V, T, Z);                      // Z' = T*Z
    wg_copy(Y, U);
    wg_copy(Z, V);
  }
  float sc = sqrtf(c);
  wg_scale(Y, Y, sc);                    // sqrt(A)
  wg_scale(Z, Z, 1.0f / sc);            // invsqrt(A)
  __syncthreads();
}

// ---------------- matrix log via inverse scaling-squaring ------------------
// Lg = log(A). A is destroyed. Uses T1..T4 as scratch. 6 LDS matrices total.
__device__ void wg_matlog(float* Lg, float* A,
                          float* T1, float* T2, float* T3, float* T4,
                          float* red) {
#pragma clang loop unroll(disable)
  for (int s = 0; s < LOG_SQ; ++s) {
    wg_sqrt_ns(T1, T2, A, T3, T4, Lg, red);   // T1 = sqrt(A)
    wg_copy(A, T1);                            // A <- A^(1/2)
  }
  // E = A - I ; log(I+E) = E - E^2/2 + E^3/3 - ...
  wg_axpbI(A, A, 1.0f, -1.0f);                 // A = E
  wg_copy(Lg, A);                              // Lg = E
  wg_copy(T1, A);                              // running power P = E
  float sign = -1.0f;
#pragma clang loop unroll(disable)
  for (int k = 2; k <= LOG_ORD; ++k) {
    wg_mm(T2, T1, A);                          // P = P*E
    wg_add_scaled(Lg, T2, sign / (float)k);
    wg_copy(T1, T2);
    sign = -sign;
  }
  wg_scale(Lg, Lg, (float)(1 << LOG_SQ));      // undo the square roots
  __syncthreads();
}

// ---------------- matrix exp via scaling-squaring --------------------------
// Eo = exp(A). A destroyed. Scratch T1,T2.
__device__ void wg_matexp(float* Eo, float* A, float* T1, float* T2) {
  wg_scale(A, A, 1.0f / (float)(1 << EXP_SC));
  wg_axpbI(Eo, A, 1.0f, 1.0f);                 // Eo = I + A
  wg_copy(T1, A);
  float fact = 1.0f;
#pragma clang loop unroll(disable)
  for (int k = 2; k <= EXP_ORD; ++k) {
    wg_mm(T2, T1, A);
    fact *= (float)k;
    wg_add_scaled(Eo, T2, 1.0f / fact);
    wg_copy(T1, T2);
  }
#pragma clang loop unroll(disable)
  for (int s = 0; s < EXP_SC; ++s) {
    wg_mm(T1, Eo, Eo);
    wg_copy(Eo, T1);
  }
  __syncthreads();
}

// ---------------- Cholesky (L lower) and lower-triangular inverse ----------
__device__ void wg_chol(float* L, const float* A, float* W) {
  wg_copy(W, A);
  for (int e = threadIdx.x; e < NN; e += NT) L[e] = 0.0f;
  __syncthreads();
#pragma clang loop unroll(disable)
  for (int k = 0; k < NMAT; ++k) {
    if (threadIdx.x == 0) L[k * NMAT + k] = sqrtf(fmaxf(W[k * NMAT + k], 1e-20f));
    __syncthreads();
    float dk = L[k * NMAT + k];
    if ((int)threadIdx.x > k && threadIdx.x < NMAT)
      L[threadIdx.x * NMAT + k] = W[threadIdx.x * NMAT + k] / dk;
    __syncthreads();
    for (int e = threadIdx.x; e < NN; e += NT) {
      int i = e >> 6, j = e & 63;
      if (i > k && j > k && j <= i)
        W[e] -= L[i * NMAT + k] * L[j * NMAT + k];
    }
    __syncthreads();
  }
}

__device__ void wg_trinv(float* Li, const float* L) {
  for (int e = threadIdx.x; e < NN; e += NT) Li[e] = 0.0f;
  __syncthreads();
  if (threadIdx.x < NMAT) {
    int c = threadIdx.x;                       // one column per thread
#pragma clang loop unroll(disable)
    for (int i = c; i < NMAT; ++i) {
      float s = (i == c) ? 1.0f : 0.0f;
      for (int j = c; j < i; ++j) s -= L[i * NMAT + j] * Li[j * NMAT + c];
      Li[i * NMAT + c] = s / L[i * NMAT + i];
    }
  }
  __syncthreads();
}

// =========================== kernels =======================================

// Stage 1: partial batch sums of X (64 workgroups x 64 matrices each).
__global__ __launch_bounds__(NT) void k_mean_partial(const float* __restrict__ X,
                                                     float* __restrict__ part) {
  const int w = blockIdx.x;
  for (int e = threadIdx.x; e < NN; e += NT) {
    float s = 0.0f;
    const float* base = X + (size_t)w * 64 * NN + e;
#pragma clang loop unroll(disable)
    for (int b = 0; b < 64; ++b) s += base[(size_t)b * NN];
    part[(size_t)w * NN + e] = s;
  }
}

// Stage 2: mean = sum/B; Ms = mean^{1/2}, Mis = mean^{-1/2} (Newton-Schulz).
__global__ __launch_bounds__(NT) void k_mean_sqrt(const float* __restrict__ part,
                                                  float* __restrict__ MsOut,
                                                  float* __restrict__ MisOut) {
  SH_MAT(A); SH_MAT(Y); SH_MAT(Z); SH_MAT(T); SH_MAT(U); SH_MAT(V);
  __shared__ float red[NT];
  for (int e = threadIdx.x; e < NN; e += NT) {
    float s = 0.0f;
#pragma clang loop unroll(disable)
    for (int p = 0; p < 64; ++p) s += part[(size_t)p * NN + e];
    A[e] = s * (1.0f / (float)BATCH);
  }
  __syncthreads();
  wg_sqrt_ns(Y, Z, A, T, U, V, red);
  for (int e = threadIdx.x; e < NN; e += NT) {
    MsOut[e]  = Y[e];
    MisOut[e] = Z[e];
  }
}

// Stage 3: per-element log(Mis*X*Mis); per-workgroup partial sums of logs.
__global__ __launch_bounds__(NT) void k_log_partial(const float* __restrict__ X,
                                                    const float* __restrict__ Mis,
                                                    float* __restrict__ logPart) {
  SH_MAT(MIS); SH_MAT(SUM);
  SH_MAT(P0); SH_MAT(P1); SH_MAT(P2); SH_MAT(P3); SH_MAT(P4); SH_MAT(P5);
  __shared__ float red[NT];
  for (int e = threadIdx.x; e < NN; e += NT) { MIS[e] = Mis[e]; SUM[e] = 0.0f; }
  __syncthreads();
  const int w = blockIdx.x;                    // 256 workgroups, 16 each
#pragma clang loop unroll(disable)
  for (int c = 0; c < 16; ++c) {
    const size_t b = (size_t)w * 16 + c;
    for (int e = threadIdx.x; e < NN; e += NT) P0[e] = X[b * NN + e];
    wg_mm(P1, MIS, P0);                        // Mis * X
    wg_mm(P0, P1, MIS);                        // Mis * X * Mis
    wg_sym(P0);
    wg_matlog(P1, P0, P2, P3, P4, P5, red);    // P1 = log(.)
    wg_add_scaled(SUM, P1, 1.0f);
    __syncthreads();
  }
  for (int e = threadIdx.x; e < NN; e += NT) logPart[(size_t)w * NN + e] = SUM[e];
}

// Stage 4: tang -> exp -> batch_mean -> chol -> Linv; also chol(bias) -> Lb.
__global__ __launch_bounds__(NT) void k_center(const float* __restrict__ logPart,
                                               const float* __restrict__ Ms,
                                               const float* __restrict__ bias,
                                               float* __restrict__ LinvOut,
                                               float* __restrict__ LbOut) {
  SH_MAT(A); SH_MAT(E); SH_MAT(T1); SH_MAT(T2); SH_MAT(T3); SH_MAT(L);
  __shared__ float red[NT];
  for (int e = threadIdx.x; e < NN; e += NT) {
    float s = 0.0f;
#pragma clang loop unroll(disable)
    for (int p = 0; p < 256; ++p) s += logPart[(size_t)p * NN + e];
    A[e] = s * (1.0f / (float)BATCH);          // tang
  }
  __syncthreads();
  wg_matexp(E, A, T1, T2);                     // E = exp(tang)
  for (int e = threadIdx.x; e < NN; e += NT) T3[e] = Ms[e];
  wg_mm(T1, T3, E);                            // Ms * E
  wg_mm(A, T1, T3);                            // batch_mean = Ms*E*Ms
  wg_sym(A);
  wg_chol(L, A, T2);                           // L = chol(batch_mean)
  wg_trinv(E, L);                              // E = L^{-1}
  for (int e = threadIdx.x; e < NN; e += NT) LinvOut[e] = E[e];
  for (int e = threadIdx.x; e < NN; e += NT) A[e] = bias[e];
  __syncthreads();
  wg_chol(L, A, T2);                           // Lb = chol(bias)
  __syncthreads();
  for (int e = threadIdx.x; e < NN; e += NT) LbOut[e] = L[e];
}

// Stage 5: Xc = Linv*X*Linv^T; logXc -> d_out; partial variance stats.
__global__ __launch_bounds__(NT) void k_whiten_log(const float* __restrict__ X,
                                                   const float* __restrict__ Linv,
                                                   float* __restrict__ outLg,
                                                   float* __restrict__ varPart) {
  SH_MAT(LI);
  SH_MAT(P0); SH_MAT(P1); SH_MAT(P2); SH_MAT(P3); SH_MAT(P4); SH_MAT(P5);
  __shared__ float red[NT];
  for (int e = threadIdx.x; e < NN; e += NT) LI[e] = Linv[e];
  __syncthreads();
  const int w = blockIdx.x;
  float accv = 0.0f;                           // only thread 0's copy matters
#pragma clang loop unroll(disable)
  for (int c = 0; c < 16; ++c) {
    const size_t b = (size_t)w * 16 + c;
    for (int e = threadIdx.x; e < NN; e += NT) P0[e] = X[b * NN + e];
    wg_mm(P1, LI, P0);                         // Linv * X
    wg_mm_nt(P0, P1, LI);                      // * Linv^T
    wg_sym(P0);
    wg_matlog(P1, P0, P2, P3, P4, P5, red);    // P1 = log(Xc)
    // fro^2 and trace reductions
    float f = 0.0f, t = 0.0f;
    for (int e = threadIdx.x; e < NN; e += NT) {
      float v = P1[e];
      f += v * v;
      if ((e >> 6) == (e & 63)) t += v;
    }
    red[threadIdx.x] = f; __syncthreads();
    for (int s = NT / 2; s > 0; s >>= 1) {
      if (threadIdx.x < (unsigned)s) red[threadIdx.x] += red[threadIdx.x + s];
      __syncthreads();
    }
    float fro2 = red[0]; __syncthreads();
    red[threadIdx.x] = t; __syncthreads();
    for (int s = NT / 2; s > 0; s >>= 1) {
      if (threadIdx.x < (unsigned)s) red[threadIdx.x] += red[threadIdx.x + s];
      __syncthreads();
    }
    float trv = red[0]; __syncthreads();
    if (threadIdx.x == 0) accv += ALPHA_F * fro2 + BETA_F * trv * trv;
    for (int e = threadIdx.x; e < NN; e += NT) outLg[b * NN + e] = P1[e];
    __syncthreads();
  }
  if (threadIdx.x == 0) varPart[w] = accv;
}

// Stage 6: var -> factor = shift / sqrt(var + eps)
__global__ __launch_bounds__(NT) void k_var(const float* __restrict__ varPart,
                                            const float* __restrict__ shift,
                                            float* __restrict__ factorOut) {
  __shared__ float red[NT];
  red[threadIdx.x] = varPart[threadIdx.x];
  __syncthreads();
  for (int s = NT / 2; s > 0; s >>= 1) {
    if (threadIdx.x < (unsigned)s) red[threadIdx.x] += red[threadIdx.x + s];
    __syncthreads();
  }
  if (threadIdx.x == 0)
    factorOut[0] = shift[0] / sqrtf(red[0] / (float)BATCH + EPS_F);
}

// Stage 7: Xs = exp(factor*logXc); out = sym(Lb * Xs * Lb^T)
__global__ __launch_bounds__(NT) void k_bias_exp(const float* __restrict__ Lb,
                                                 const float* __restrict__ factor,
                                                 float* __restrict__ out) {
  SH_MAT(LB); SH_MAT(P0); SH_MAT(P1); SH_MAT(P2); SH_MAT(P3);
  for (int e = threadIdx.x; e < NN; e += NT) LB[e] = Lb[e];
  const float f = factor[0];
  __syncthreads();
  const int w = blockIdx.x;
#pragma clang loop unroll(disable)
  for (int c = 0; c < 16; ++c) {
    const size_t b = (size_t)w * 16 + c;
    for (int e = threadIdx.x; e < NN; e += NT) P0[e] = f * out[b * NN + e];
    __syncthreads();
    wg_matexp(P1, P0, P2, P3);                 // Xs = exp(f*logXc)
    wg_mm(P0, LB, P1);                         // Lb * Xs
    wg_mm_nt(P1, P0, LB);                      // * Lb^T
    wg_sym(P1);
    for (int e = threadIdx.x; e < NN; e += NT) out[b * NN + e] = P1[e];
    __syncthreads();
  }
}

// =========================== host entry ====================================
extern "C" void kernel_launch(void* const* d_in, const int* in_sizes, int n_in,
                              void* d_out, int out_size, void* d_ws, size_t ws_size,
                              hipStream_t stream) {
  (void)in_sizes; (void)n_in; (void)out_size; (void)ws_size;
  const float* X     = (const float*)d_in[0];   // [4096,64,64]
  const float* bias  = (const float*)d_in[1];   // [1,64,64]
  const float* shift = (const float*)d_in[2];   // [1]
  float* out = (float*)d_out;                   // [4096,64,64]
  float* ws  = (float*)d_ws;

  float* meanPart = ws;                         // 64*4096
  float* Ms       = meanPart + 64 * NN;         // 4096
  float* Mis      = Ms + NN;                    // 4096
  float* Linv     = Mis + NN;                   // 4096
  float* Lb       = Linv + NN;                  // 4096
  float* factor   = Lb + NN;                    // 1
  float* varPart  = factor + 1;                 // 256
  float* logPart  = varPart + 256;              // 256*4096

  k_mean_partial<<<64,  NT, 0, stream>>>(X, meanPart);
  k_mean_sqrt   <<<1,   NT, 0, stream>>>(meanPart, Ms, Mis);
  k_log_partial <<<256, NT, 0, stream>>>(X, Mis, logPart);
  k_center      <<<1,   NT, 0, stream>>>(logPart, Ms, bias, Linv, Lb);
  k_whiten_log  <<<256, NT, 0, stream>>>(X, Linv, out, varPart);
  k_var         <<<1,   NT, 0, stream>>>(varPart, shift, factor);
  k_bias_exp    <<<256, NT, 0, stream>>>(Lb, factor, out);
}